// Att_cat_norm_42984032698904
// MI455X (gfx1250) — compile-verified
//
#include <hip/hip_runtime.h>
#include <math.h>

// ---------------------------------------------------------------------------
// Problem constants (B,N,D from reference)
// ---------------------------------------------------------------------------
#define BB 4
#define NN 512
#define DD 128
#define LEAKY 0.01f
#define LN_EPS 1e-5f

typedef float __attribute__((ext_vector_type(2))) v2f;
typedef float __attribute__((ext_vector_type(4))) v4f;
typedef float __attribute__((ext_vector_type(8))) v8f;

// ---------------------------------------------------------------------------
// Kernel 1: LayerNorm + projection to (sq, sk) via V_WMMA_F32_16X16X4_F32.
// One wave (32 threads) handles a tile of 16 rows of the (B*N, D) matrix.
// S(16x2) = LN(rows)(16x128) x W^T(128x2), accumulated over 32 K=4 steps.
// B-fragment is built branchlessly (pointer select + lane mask) so EXEC stays
// all-ones through the whole WMMA loop — no saveexec churn.
// ---------------------------------------------------------------------------
__global__ __launch_bounds__(32)
void scores_wmma_kernel(const float* __restrict__ emb,
                        const float* __restrict__ att_w,   // (1, 2D): [wq | wk]
                        const float* __restrict__ gamma,
                        const float* __restrict__ beta,
                        float* __restrict__ sq,            // (B*N)
                        float* __restrict__ sk)            // (B*N)
{
    const int lane  = threadIdx.x;          // 0..31
    const int sub   = lane & 15;            // row (A) / col (B) within tile
    const int hi    = lane >> 4;            // half-wave select
    const int rbase = blockIdx.x * 16;      // first global row of this tile

    // ---- LayerNorm statistics: each lane sums one half (64 elems) of row `sub`,
    //      combined with its partner lane (lane ^ 16).
    const float* xrow = emb + (size_t)(rbase + sub) * DD;
    float s = 0.f, s2 = 0.f;
    const int k0 = hi * (DD / 2);
    #pragma unroll 8
    for (int k = 0; k < DD / 2; ++k) {
        float v = xrow[k0 + k];
        s += v; s2 += v * v;
    }
    s  += __shfl_xor(s, 16, 32);
    s2 += __shfl_xor(s2, 16, 32);
    const float mu   = s * (1.0f / DD);
    const float var  = s2 * (1.0f / DD) - mu * mu;
    const float rstd = rsqrtf(var + LN_EPS);

    // ---- Branchless B-operand source: column 0 = wq, column 1 = wk, rest 0.
    // All lanes issue the same-shaped load (lanes >=2 alias wk's cacheline)
    // and the result is zeroed by `msk` — no EXEC manipulation.
    const float* wsel = (sub == 0) ? att_w : (att_w + DD);
    const float  msk  = (sub < 2) ? 1.0f : 0.0f;

    // ---- WMMA accumulation over K = 128 in chunks of 4.
    // A (16x4 f32): lanes 0-15 hold {K=0,K=1}, lanes 16-31 hold {K=2,K=3}.
    // B (4x16 f32): mirrored — lane sub = column n; this lane-half holds
    //               {K=2*hi, K=2*hi+1}.
    v8f c = {};
    #pragma unroll 4
    for (int kk = 0; kk < DD / 4; ++kk) {
        const int kb = kk * 4 + hi * 2;

        const v2f xv = *(const v2f*)(xrow  + kb);
        const v2f gv = *(const v2f*)(gamma + kb);
        const v2f bv = *(const v2f*)(beta  + kb);
        const v2f wv = *(const v2f*)(wsel  + kb);

        v2f a;
        a.x = (xv.x - mu) * rstd * gv.x + bv.x;
        a.y = (xv.y - mu) * rstd * gv.y + bv.y;

        v2f b;
        b.x = wv.x * msk;
        b.y = wv.y * msk;

        c = __builtin_amdgcn_wmma_f32_16x16x4_f32(
                /*neg_a=*/false, a, /*neg_b=*/false, b,
                /*c_mod=*/(short)0, c, /*reuse_a=*/false, /*reuse_b=*/false);
    }

    // ---- Extract column 0 (sq) / column 1 (sk) from the C/D layout:
    // VGPR v: lanes 0-15 -> M=v, N=lane ; lanes 16-31 -> M=v+8, N=lane-16.
    if (lane == 0) {
        #pragma unroll
        for (int v = 0; v < 8; ++v) sq[rbase + v] = c[v];
    } else if (lane == 16) {
        #pragma unroll
        for (int v = 0; v < 8; ++v) sq[rbase + 8 + v] = c[v];
    } else if (lane == 1) {
        #pragma unroll
        for (int v = 0; v < 8; ++v) sk[rbase + v] = c[v];
    } else if (lane == 17) {
        #pragma unroll
        for (int v = 0; v < 8; ++v) sk[rbase + 8 + v] = c[v];
    }
}

// ---------------------------------------------------------------------------
// Kernel 2: logits = sq[i] + sk[j] + b, LeakyReLU, softmax over j.
// One wave per (b,i) row; 16 logits per lane in registers; shfl reductions.
// ---------------------------------------------------------------------------
__global__ __launch_bounds__(32)
void softmax_kernel(const float* __restrict__ sq,
                    const float* __restrict__ sk,
                    const float* __restrict__ att_b,
                    float* __restrict__ alphas)            // (B*N, N)
{
    const int row  = blockIdx.x;          // b*N + i
    const int b    = row >> 9;            // N = 512
    const int lane = threadIdx.x;

    const float q    = sq[row];
    const float bias = att_b[0];
    const float* skb = sk + b * NN;

    float l[16];
    float m = -INFINITY;
    #pragma unroll
    for (int t = 0; t < 16; ++t) {
        float v = q + skb[t * 32 + lane] + bias;
        v = (v >= 0.0f) ? v : LEAKY * v;
        l[t] = v;
        m = fmaxf(m, v);
    }
    #pragma unroll
    for (int off = 16; off > 0; off >>= 1)
        m = fmaxf(m, __shfl_xor(m, off, 32));

    float ssum = 0.f;
    #pragma unroll
    for (int t = 0; t < 16; ++t) {
        float e = __expf(l[t] - m);
        l[t] = e;
        ssum += e;
    }
    #pragma unroll
    for (int off = 16; off > 0; off >>= 1)
        ssum += __shfl_xor(ssum, off, 32);

    const float inv = 1.0f / ssum;
    float* out = alphas + (size_t)row * NN;
    #pragma unroll
    for (int t = 0; t < 16; ++t)
        out[t * 32 + lane] = l[t] * inv;
}

// ---------------------------------------------------------------------------
// Kernel 3: value[b,i,j,:] = emb[b,i,:] * emb[b,j,:]  (536 MB of stores).
// One 256-thread block per (b,i). Thread owns a fixed float4 of emb[b,i] in
// registers; streams j with coalesced b128 loads (L2-resident: 256 KB/batch)
// and non-temporal b128 stores (bypass L2 dirtying for the huge output).
// Each wave writes one full contiguous 512 B j-row per iteration.
// ---------------------------------------------------------------------------
__global__ __launch_bounds__(256)
void value_outer_kernel(const float* __restrict__ emb,
                        float* __restrict__ value)         // (B*N, N, D)
{
    const int blk = blockIdx.x;           // b*N + i
    const int b   = blk >> 9;
    const int tid = threadIdx.x;
    const int d4  = tid & 31;             // float4 index within a D=128 row
    const int jl  = tid >> 5;             // 0..7: j sub-row handled by this wave

    const v4f xi = ((const v4f*)(emb + (size_t)blk * DD))[d4];
    const float* embB = emb + (size_t)b * NN * DD;
    float* out = value + (size_t)blk * NN * DD;

    #pragma unroll 4
    for (int jj = 0; jj < NN / 8; ++jj) {
        const int j = jj * 8 + jl;
        const v4f xj = ((const v4f*)(embB + (size_t)j * DD))[d4];
        const v4f r = xi * xj;
        __builtin_nontemporal_store(r, (v4f*)(out + (size_t)j * DD) + d4);
    }
}

// ---------------------------------------------------------------------------
// Entry point
// ---------------------------------------------------------------------------
extern "C" void kernel_launch(void* const* d_in, const int* in_sizes, int n_in,
                              void* d_out, int out_size, void* d_ws, size_t ws_size,
                              hipStream_t stream) {
    const float* emb   = (const float*)d_in[0];   // (B,N,D)
    const float* att_w = (const float*)d_in[1];   // (1,2D)
    const float* att_b = (const float*)d_in[2];   // (1,)
    const float* gamma = (const float*)d_in[3];   // (D,)
    const float* beta  = (const float*)d_in[4];   // (D,)

    float* alphas = (float*)d_out;                             // (B,N,N,1)
    float* value  = alphas + (size_t)BB * NN * NN;             // (B,N,N,D)

    float* sq = (float*)d_ws;                                  // (B*N)
    float* sk = sq + BB * NN;                                  // (B*N)

    const int rows = BB * NN;                                  // 2048

    // 1) LN + WMMA projection -> sq, sk
    scores_wmma_kernel<<<rows / 16, 32, 0, stream>>>(emb, att_w, gamma, beta, sq, sk);
    // 2) leaky + softmax -> alphas
    softmax_kernel<<<rows, 32, 0, stream>>>(sq, sk, att_b, alphas);
    // 3) outer-product value tensor (independent; bandwidth-bound)
    value_outer_kernel<<<rows, 256, 0, stream>>>(emb, value);
}